// KernelConv_28028956574292
// MI455X (gfx1250) — compile-verified
//
#include <hip/hip_runtime.h>
#include <stdint.h>

#define B_N 4
#define C_N 3
#define H_N 384
#define W_N 384
#define K_N 15
#define P_N 7
#define HWSZ (H_N * W_N)

#define TX 64              // output columns per block
#define TY 16              // output rows per block
#define VEC 4              // outputs per thread in x (16 threads.x * 4)
#define LROWS (TY + K_N - 1)  // 30 data rows incl. halo
#define LCOLS 80              // data cols incl. halo, 16B-aligned windows

typedef float v4f __attribute__((ext_vector_type(4)));

// Exact parameter type of the gfx1250 async-load builtins (from the compiler
// diagnostic): gcc-style int4 vector pointers, AS1 (global) / AS3 (LDS).
typedef int v4i_raw __attribute__((__vector_size__(16)));
typedef __attribute__((address_space(1))) v4i_raw g_v4i;
typedef __attribute__((address_space(3))) v4i_raw l_v4i;

#if __has_builtin(__builtin_amdgcn_global_load_async_to_lds_b128)
#define HAVE_ASYNC 1
#else
#define HAVE_ASYNC 0
#endif

__device__ __forceinline__ void load_tile_b128(const float* __restrict__ g, float* l) {
#if HAVE_ASYNC
  // (global src, lds dst, imm offset, imm cpol) -> global_load_async_to_lds_b128
  __builtin_amdgcn_global_load_async_to_lds_b128(
      (g_v4i*)(uintptr_t)g,
      (l_v4i*)(uint32_t)(uintptr_t)l,   // low 32 bits of generic ptr == LDS offset
      0, 0);
#else
  *(v4f*)l = *(const v4f*)g;
#endif
}

__device__ __forceinline__ void wait_async_all() {
#if HAVE_ASYNC
#if __has_builtin(__builtin_amdgcn_s_wait_asynccnt)
  __builtin_amdgcn_s_wait_asynccnt(0);
#else
  asm volatile("s_wait_asynccnt 0" ::: "memory");
#endif
#endif
}

__global__ __launch_bounds__(256)
void svconv15_kernel(const float* __restrict__ data,
                     const float* __restrict__ kern,
                     float* __restrict__ out) {
  __shared__ __align__(16) float tile[C_N][LROWS][LCOLS];  // 28.8 KB

  const int tx = threadIdx.x;            // 0..15
  const int ty = threadIdx.y;            // 0..15
  const int tid = ty * 16 + tx;
  const int xBase = blockIdx.x * TX;
  const int yBase = blockIdx.y * TY;
  const int b = blockIdx.z;

  // ---- stage data tile (+halo, zero-padded) into LDS with async b128 loads ----
  // tile covers global rows [yBase-7, yBase+22], cols [xBase-8, xBase+71]
  const int NV = C_N * LROWS * (LCOLS / 4);  // 1800 float4 groups
  for (int v = tid; v < NV; v += 256) {
    int c   = v / (LROWS * (LCOLS / 4));
    int rem = v - c * (LROWS * (LCOLS / 4));
    int lr  = rem / (LCOLS / 4);
    int lc  = (rem - lr * (LCOLS / 4)) * 4;
    int gr  = yBase - P_N + lr;
    int gc  = xBase - (P_N + 1) + lc;      // multiple of 4 -> float4 fully in/out
    float* dst = &tile[c][lr][lc];
    if ((unsigned)gr < (unsigned)H_N && (unsigned)gc < (unsigned)W_N) {
      const float* src = data + ((size_t)(b * C_N + c) * H_N + gr) * W_N + gc;
      load_tile_b128(src, dst);
    } else {
      *(v4f*)dst = (v4f)0.0f;
    }
  }
  wait_async_all();
  __syncthreads();

  // ---- accumulate: out[y][x] = sum_{i,j} kern[b,i,j,y,x] * data[y+7-i, x+7-j] ----
  const int y  = yBase + ty;
  const int x0 = xBase + tx * VEC;

  float acc[C_N][VEC];
#pragma unroll
  for (int c = 0; c < C_N; ++c)
#pragma unroll
    for (int t = 0; t < VEC; ++t) acc[c][t] = 0.0f;

  const float* kbase = kern + (size_t)b * (K_N * K_N) * HWSZ + (size_t)y * W_N + x0;

  for (int i = 0; i < K_N; ++i) {
    const int lr = ty + (K_N - 1) - i;     // LDS row holding data row y+7-i

    // 20-float register window per channel: dr[c][m] = data col (x0 - 8 + m)
    float dr[C_N][20];
#pragma unroll
    for (int c = 0; c < C_N; ++c) {
#pragma unroll
      for (int k = 0; k < 5; ++k) {
        v4f q = *(const v4f*)&tile[c][lr][tx * VEC + 4 * k];
        dr[c][4 * k + 0] = q.x;
        dr[c][4 * k + 1] = q.y;
        dr[c][4 * k + 2] = q.z;
        dr[c][4 * k + 3] = q.w;
      }
    }

    const float* wrow = kbase + (size_t)i * K_N * HWSZ;
#pragma unroll
    for (int j = 0; j < K_N; ++j) {
      // single-use streaming weights: non-temporal b128, 256B/row coalesced
      v4f w = __builtin_nontemporal_load((const v4f*)(wrow + (size_t)j * HWSZ));
#pragma unroll
      for (int c = 0; c < C_N; ++c) {
#pragma unroll
        for (int t = 0; t < VEC; ++t) {
          // data col = x0 + t + 7 - j  ->  window index m = t + 15 - j (const)
          acc[c][t] = __builtin_fmaf(w[t], dr[c][t + K_N - j], acc[c][t]);
        }
      }
    }
  }

  // ---- store (coalesced float4) ----
#pragma unroll
  for (int c = 0; c < C_N; ++c) {
    v4f r;
    r.x = acc[c][0]; r.y = acc[c][1]; r.z = acc[c][2]; r.w = acc[c][3];
    *(v4f*)(out + ((size_t)(b * C_N + c) * H_N + y) * W_N + x0) = r;
  }
}

extern "C" void kernel_launch(void* const* d_in, const int* in_sizes, int n_in,
                              void* d_out, int out_size, void* d_ws, size_t ws_size,
                              hipStream_t stream) {
  const float* data = (const float*)d_in[0];  // [4,3,384,384] f32
  const float* kern = (const float*)d_in[1];  // [4,15,15,384,384] f32
  float* out = (float*)d_out;                 // [4,3,384,384] f32

  dim3 grid(W_N / TX, H_N / TY, B_N);   // (6, 24, 4) = 576 blocks
  dim3 block(16, 16, 1);                // 256 threads = 8 wave32
  hipLaunchKernelGGL(svconv15_kernel, grid, block, 0, stream, data, kern, out);
}